// UNFLayer_57492432224267
// MI455X (gfx1250) — compile-verified
//
#include <hip/hip_runtime.h>

typedef __attribute__((ext_vector_type(2))) float v2f;
typedef __attribute__((ext_vector_type(8))) float v8f;

#define NN 1024
#define CH 16

// workspace float offsets
#define WS_S1 0
#define WS_T1 16384
#define WS_S2 32768
#define WS_T2 49152
#define WS_A0 65536
#define WS_B0 81920
#define WS_A1 98304
#define WS_B1 114688
#define WS_C0 131072
#define WS_C1 131088
#define WS_C2 131104
#define WS_C3 131120

// output float offsets (out0, out1, out2, out3 concatenated)
#define OUT1_OFF (NN*NN*CH)
#define OUT2_OFF (2*NN*NN*CH)
#define OUT3_OFF (2*NN*NN*CH + NN*CH)

// ---------------------------------------------------------------------------
// Kernel 0: zero the atomic accumulation buffers S1, S2 (16384 floats each)
// ---------------------------------------------------------------------------
__global__ __launch_bounds__(256) void k_zero(float* __restrict__ ws) {
    int i = blockIdx.x * 256 + threadIdx.x;   // grid 64 -> i in [0,16384)
    ws[WS_S1 + i] = 0.0f;
    ws[WS_S2 + i] = 0.0f;
}

// ---------------------------------------------------------------------------
// Kernel 1: fused row+column reduction, single read of W1/W2.
//  block = (blockIdx.x: 8-row tile, blockIdx.y: tensor), 256 threads.
//  T[u,c] = sum_v W[u,v,c] via LDS reduce; S[v,c] partials via f32 atomics.
// ---------------------------------------------------------------------------
__global__ __launch_bounds__(256) void k_reduce(const float* __restrict__ W1,
                                                const float* __restrict__ W2,
                                                float* __restrict__ ws) {
    const int tensor = blockIdx.y;
    const float* __restrict__ W = tensor ? W2 : W1;
    float* __restrict__ S = ws + (tensor ? WS_S2 : WS_S1);
    float* __restrict__ T = ws + (tensor ? WS_T2 : WS_T1);
    const int t  = threadIdx.x;
    const int r0 = blockIdx.x * 8;
    __shared__ float red[256];

    float sacc[64];
#pragma unroll
    for (int k = 0; k < 64; ++k) sacc[k] = 0.0f;

    for (int r = 0; r < 8; ++r) {
        const float* __restrict__ row = W + (size_t)(r0 + r) * (NN * CH);
        float tacc = 0.0f;
#pragma unroll
        for (int k = 0; k < 64; ++k) {           // 256 threads x 64 = 16384 floats
            float x = row[t + 256 * k];          // coalesced 1KB per instruction
            sacc[k] += x;                        // column (v,c) partials
            tacc += x;                           // row partial, channel = t & 15
        }
        red[t] = tacc;
        __syncthreads();
        if (t < 16) {
            float s = 0.0f;
#pragma unroll
            for (int j = 0; j < 16; ++j) s += red[t + 16 * j];
            T[(r0 + r) * 16 + t] = s;
        }
        __syncthreads();
    }
#pragma unroll
    for (int k = 0; k < 64; ++k) atomicAdd(&S[t + 256 * k], sacc[k]);
}

// ---------------------------------------------------------------------------
// Kernel 2: global scalars U1,U2,sb1,sb2 and constant vectors c0..c3 (1 block)
// ---------------------------------------------------------------------------
__global__ __launch_bounds__(256) void k_finalize(
        const float* __restrict__ b1, const float* __restrict__ b2,
        const float* __restrict__ th0, const float* __restrict__ th1,
        const float* __restrict__ th2, const float* __restrict__ th3,
        const float* __restrict__ bias0, const float* __restrict__ bias1,
        const float* __restrict__ bias2, const float* __restrict__ bias3,
        float* __restrict__ ws) {
    const int t = threadIdx.x;
    __shared__ float red[256];
    __shared__ float U[4][16];     // U1, U2, sb1, sb2

    const int c = t & 15, g = t >> 4;
    for (int s = 0; s < 4; ++s) {
        const float* p = (s == 0) ? (ws + WS_T1) : (s == 1) ? (ws + WS_T2)
                        : (s == 2) ? b1 : b2;
        float acc = 0.0f;
        for (int k = 0; k < 64; ++k) acc += p[(g + 16 * k) * 16 + c];
        red[t] = acc;
        __syncthreads();
        if (t < 16) {
            float s2 = 0.0f;
#pragma unroll
            for (int j = 0; j < 16; ++j) s2 += red[t + 16 * j];
            U[s][t] = s2;
        }
        __syncthreads();
    }
    if (t < 64) {
        const int grp = t >> 4, o = t & 15;
        float acc = 0.0f;
        if (grp == 0) {
            for (int cc = 0; cc < 16; ++cc) {
                int i = cc * 16 + o;
                acc += U[0][cc]*th0[3*256+i] + U[1][cc]*th0[5*256+i]
                     + U[2][cc]*th0[7*256+i] + U[3][cc]*th0[8*256+i];
            }
            ws[WS_C0 + o] = acc + bias0[o];
        } else if (grp == 1) {
            for (int cc = 0; cc < 16; ++cc) {
                int i = cc * 16 + o;
                acc += U[0][cc]*th1[1*256+i] + U[1][cc]*th1[5*256+i]
                     + U[2][cc]*th1[7*256+i] + U[3][cc]*th1[9*256+i];
            }
            ws[WS_C1 + o] = acc + bias1[o];
        } else if (grp == 2) {
            for (int cc = 0; cc < 16; ++cc) {
                int i = cc * 16 + o;
                acc += U[0][cc]*th2[1*256+i] + U[1][cc]*th2[3*256+i]
                     + U[2][cc]*th2[5*256+i] + U[3][cc]*th2[6*256+i];
            }
            ws[WS_C2 + o] = acc + bias2[o];
        } else {
            for (int cc = 0; cc < 16; ++cc) {
                int i = cc * 16 + o;
                acc += U[0][cc]*th3[0*256+i] + U[1][cc]*th3[2*256+i]
                     + U[2][cc]*th3[3*256+i] + U[3][cc]*th3[5*256+i];
            }
            ws[WS_C3 + o] = acc + bias3[o];
        }
    }
}

// ---------------------------------------------------------------------------
// Kernel 3: per-row vectors A0,B0,A1,B1 and small outputs out2/out3.
//  grid 64 x 256 threads: thread = (row r = bx*16 + t/16, out channel o = t%16)
// ---------------------------------------------------------------------------
__global__ __launch_bounds__(256) void k_prep(
        const float* __restrict__ b1, const float* __restrict__ b2,
        const float* __restrict__ th0, const float* __restrict__ th1,
        const float* __restrict__ th2, const float* __restrict__ th3,
        float* __restrict__ ws, float* __restrict__ out) {
    const int t = threadIdx.x;
    const int r = blockIdx.x * 16 + (t >> 4);
    const int o = t & 15;
    const float* S1 = ws + WS_S1 + r * 16;
    const float* T1 = ws + WS_T1 + r * 16;
    const float* S2 = ws + WS_S2 + r * 16;
    const float* T2 = ws + WS_T2 + r * 16;
    const float* X1 = b1 + r * 16;
    const float* X2 = b2 + r * 16;

    float a0 = 0.f, b0 = 0.f, a1 = 0.f, b1v = 0.f, o2 = 0.f, o3 = 0.f;
#pragma unroll
    for (int c = 0; c < 16; ++c) {
        const float s1 = S1[c], t1 = T1[c], s2 = S2[c], t2 = T2[c];
        const float x1 = X1[c], x2 = X2[c];
        const int i = c * 16 + o;
        a0  += t1 * th0[1*256+i];
        b0  += s1 * th0[2*256+i] + t2 * th0[4*256+i] + x1 * th0[6*256+i];
        a1  += s1 * th1[0*256+i] + t2 * th1[3*256+i] + x1 * th1[6*256+i];
        b1v += s2 * th1[4*256+i] + x2 * th1[8*256+i];
        o2  += s1 * th2[0*256+i] + t2 * th2[2*256+i] + x1 * th2[4*256+i];
        o3  += s2 * th3[1*256+i] + x2 * th3[4*256+i];
    }
    ws[WS_A0 + r * 16 + o] = a0;
    ws[WS_B0 + r * 16 + o] = b0;
    ws[WS_A1 + r * 16 + o] = a1;
    ws[WS_B1 + r * 16 + o] = b1v;
    out[OUT2_OFF + r * 16 + o] = o2 + ws[WS_C2 + o];
    out[OUT3_OFF + r * 16 + o] = o3 + ws[WS_C3 + o];
}

// ---------------------------------------------------------------------------
// Kernel 4: main pass. out{0,1}[u,v,:] = W[u,v,:] @ theta + A[u,:] + B[v,:] + c
//  One block per (u, tensor); 8 waves; each wave does 16-position tiles:
//  D(16x16) = A(16pos x 4ch) x B(4ch x 16out) chained over 4 K-chunks,
//  with C pre-loaded with the additive rank-1 parts.
// ---------------------------------------------------------------------------
__global__ __launch_bounds__(256) void k_main(
        const float* __restrict__ W1, const float* __restrict__ W2,
        const float* __restrict__ th0, const float* __restrict__ th1,
        const float* __restrict__ ws, float* __restrict__ out) {
    const int tensor = blockIdx.y;
    const float* __restrict__ X   = tensor ? W2 : W1;
    const float* __restrict__ th  = tensor ? (th1 + 2 * 256) : th0;   // theta1[2] / theta0[0]
    const float* __restrict__ adA = ws + (tensor ? WS_A1 : WS_A0);
    const float* __restrict__ adB = ws + (tensor ? WS_B1 : WS_B0);
    const float* __restrict__ cst = ws + (tensor ? WS_C1 : WS_C0);
    float* __restrict__ O = out + (tensor ? OUT1_OFF : 0);

    const int u    = blockIdx.x;
    const int lane = threadIdx.x & 31;
    const int wave = threadIdx.x >> 5;
    const int n    = lane & 15;     // N index (out channel) / M index for A
    const int hi   = lane >> 4;     // lane-half selects K (A/B) or M-half (C/D)

    // B (theta) registers: chunk q, element (K = 4q + 2*hi + g, N = n)
    v2f b[4];
#pragma unroll
    for (int q = 0; q < 4; ++q) {
        const int c0 = 4 * q + 2 * hi;
        b[q].x = th[c0 * 16 + n];
        b[q].y = th[(c0 + 1) * 16 + n];
    }
    const float aconst = adA[u * 16 + n] + cst[n];   // per-N, uniform over M
    const float* __restrict__ Xrow = X + (size_t)u * (NN * CH);
    float* __restrict__ Orow = O + (size_t)u * (NN * CH);

    for (int v0 = wave * 16; v0 < NN; v0 += 8 * 16) {
        // C/D layout: vgpr g, lanes0-15 -> M=g, lanes16-31 -> M=g+8, N=n
        v8f c;
#pragma unroll
        for (int g = 0; g < 8; ++g)
            c[g] = adB[(v0 + g + 8 * hi) * 16 + n] + aconst;

        // A layout: lane -> M = n (position in tile), K = 4q + 2*hi + {0,1}
        const float* Ap = Xrow + (size_t)(v0 + n) * 16 + 2 * hi;
#pragma unroll
        for (int q = 0; q < 4; ++q) {
            v2f a = *(const v2f*)(Ap + 4 * q);
            c = __builtin_amdgcn_wmma_f32_16x16x4_f32(
                    false, a, false, b[q], (short)0, c, false, false);
        }
        // streaming stores (keep W-tensors resident in L2 for this pass)
        float* Op = Orow + (size_t)(v0 + 8 * hi) * 16 + n;
#pragma unroll
        for (int g = 0; g < 8; ++g)
            __builtin_nontemporal_store(c[g], Op + g * 16);
    }
}

// ---------------------------------------------------------------------------
extern "C" void kernel_launch(void* const* d_in, const int* in_sizes, int n_in,
                              void* d_out, int out_size, void* d_ws, size_t ws_size,
                              hipStream_t stream) {
    const float* W1 = (const float*)d_in[0];
    const float* W2 = (const float*)d_in[1];
    const float* b1 = (const float*)d_in[2];
    const float* b2 = (const float*)d_in[3];
    const float* th0 = (const float*)d_in[4];
    const float* bz0 = (const float*)d_in[5];
    const float* th1 = (const float*)d_in[6];
    const float* bz1 = (const float*)d_in[7];
    const float* th2 = (const float*)d_in[8];
    const float* bz2 = (const float*)d_in[9];
    const float* th3 = (const float*)d_in[10];
    const float* bz3 = (const float*)d_in[11];
    float* ws  = (float*)d_ws;
    float* out = (float*)d_out;

    k_zero<<<dim3(64), 256, 0, stream>>>(ws);
    k_reduce<<<dim3(128, 2), 256, 0, stream>>>(W1, W2, ws);
    k_finalize<<<dim3(1), 256, 0, stream>>>(b1, b2, th0, th1, th2, th3,
                                            bz0, bz1, bz2, bz3, ws);
    k_prep<<<dim3(64), 256, 0, stream>>>(b1, b2, th0, th1, th2, th3, ws, out);
    k_main<<<dim3(1024, 2), 256, 0, stream>>>(W1, W2, th0, th1, ws, out);
}